// Quantizer_6150393168136
// MI455X (gfx1250) — compile-verified
//
#include <hip/hip_runtime.h>

typedef __attribute__((ext_vector_type(2))) float v2f;
typedef __attribute__((ext_vector_type(8))) float v8f;

#define EMB_DIM   32
#define NUM_EMB   1024
#define ROW_PAD   34   // 32 floats + 2 pad -> WMMA-pattern LDS reads hit all 64 banks once
#define LDS_FLOATS (NUM_EMB * ROW_PAD + NUM_EMB)  // padded codebook + eNorm
// commitment_loss * 0.1 + embedding_loss * 0.1 (numerically equal values)
#define LOSS_SCALE 0.2f

#if defined(__has_builtin)
#  if __has_builtin(__builtin_amdgcn_global_load_async_to_lds_b128)
#    define HAVE_ASYNC_LDS 1
#  endif
#  if __has_builtin(__builtin_amdgcn_s_wait_asynccnt)
#    define HAVE_WAIT_ASYNC 1
#  endif
#endif

#if defined(HAVE_ASYNC_LDS)
// Builtin signature (recovered from clang diagnostics):
//   void __builtin_amdgcn_global_load_async_to_lds_b128(
//       int __vector(4) addrspace(1)* src, int __vector(4) addrspace(3)* dst,
//       imm int offset, imm int cpol)
typedef int v4i_vs __attribute__((__vector_size__(16)));
typedef __attribute__((address_space(1))) v4i_vs global_v4i;
typedef __attribute__((address_space(3))) v4i_vs lds_v4i;
#endif

// ---------------------------------------------------------------------------
// Fused kernel: stage codebook in LDS -> eNorm -> f32 WMMA distance GEMM ->
// running argmin. One wave32 handles TWO 16-row blocks (two independent WMMA
// accumulator chains per B chunk).
//
// A-matrix (16x4 f32) VGPR layout (ISA 7.12.2):
//   lanes 0-15  : M = lane,    v0 -> K=0, v1 -> K=1
//   lanes 16-31 : M = lane-16, v0 -> K=2, v1 -> K=3
// B-matrix (4x16) mirrored; C/D 16x16: VGPR r -> (M=r | r+8, N=lane%16).
// ---------------------------------------------------------------------------
__global__ __launch_bounds__(256) void vq_argmin_kernel(const float* __restrict__ h,
                                                        const float* __restrict__ emb,
                                                        int* __restrict__ idxOut) {
    extern __shared__ float lds[];
    float* __restrict__ embLds   = lds;                      // NUM_EMB * ROW_PAD
    float* __restrict__ eNormLds = lds + NUM_EMB * ROW_PAD;  // NUM_EMB

    const int tid = threadIdx.x;

    // ---- Stage the 128 KB codebook into LDS (once per workgroup) ----
#pragma unroll
    for (int j = 0; j < (NUM_EMB * EMB_DIM / 4) / 256; ++j) {  // 32 float4 per thread
        const int i   = tid + j * 256;   // float4 index into emb
        const int row = i >> 3;          // 8 float4 per 32-float row
        const int col = (i & 7) << 2;
        float* dst = &embLds[row * ROW_PAD + col];
#if defined(HAVE_ASYNC_LDS)
        __builtin_amdgcn_global_load_async_to_lds_b128(
            (global_v4i*)(emb + (size_t)i * 4),
            (lds_v4i*)dst, 0, 0);
#else
        *(float4*)dst = ((const float4*)emb)[i];
#endif
    }
#if defined(HAVE_ASYNC_LDS)
#  if defined(HAVE_WAIT_ASYNC)
    __builtin_amdgcn_s_wait_asynccnt(0);
#  else
    asm volatile("s_wait_asynccnt 0" ::: "memory");
#  endif
#endif
    __syncthreads();

    // ---- Per-embedding squared norms into LDS ----
#pragma unroll
    for (int j = 0; j < NUM_EMB / 256; ++j) {
        const int n = tid + j * 256;
        const float* r = &embLds[n * ROW_PAD];
        float s = 0.0f;
#pragma unroll
        for (int d = 0; d < EMB_DIM; ++d) s += r[d] * r[d];
        eNormLds[n] = s;
    }
    __syncthreads();

    // ---- WMMA distance GEMM + argmin ----
    const int lane     = tid & 31;
    const int wave     = tid >> 5;
    const int rowBase0 = (blockIdx.x * 8 + wave) << 5;  // 32 rows per wave
    const int rowBase1 = rowBase0 + 16;
    const int m        = lane & 15;   // A row / B,C column
    const int hi       = lane >> 4;   // K-pair select {0,1} vs {2,3}

    // Resident A fragments for both row blocks (16 v2f = 32 VGPRs)
    v2f a0[8], a1[8];
    const float* ar0 = h + (size_t)(rowBase0 + m) * EMB_DIM + 2 * hi;
    const float* ar1 = h + (size_t)(rowBase1 + m) * EMB_DIM + 2 * hi;
#pragma unroll
    for (int k = 0; k < 8; ++k) {
        float2 t0 = *(const float2*)(ar0 + 4 * k);
        float2 t1 = *(const float2*)(ar1 + 4 * k);
        a0[k].x = t0.x; a0[k].y = t0.y;
        a1[k].x = t1.x; a1[k].y = t1.y;
    }

    float minv0[8], minv1[8];
    int   mini0[8], mini1[8];
#pragma unroll
    for (int r = 0; r < 8; ++r) {
        minv0[r] = __builtin_inff(); mini0[r] = 0;
        minv1[r] = __builtin_inff(); mini1[r] = 0;
    }

    const float* bbase = embLds + m * ROW_PAD + 2 * hi;
    for (int colBase = 0; colBase < NUM_EMB; colBase += 16) {
        const float* brow = bbase + colBase * ROW_PAD;
        v8f acc0 = {0.f,0.f,0.f,0.f,0.f,0.f,0.f,0.f};
        v8f acc1 = {0.f,0.f,0.f,0.f,0.f,0.f,0.f,0.f};
#pragma unroll
        for (int k = 0; k < 8; ++k) {
            float2 t = *(const float2*)(brow + 4 * k);  // ds_load_b64, bank-conflict-free
            v2f b; b.x = t.x; b.y = t.y;
            // Two independent accumulator chains share each B chunk.
            acc0 = __builtin_amdgcn_wmma_f32_16x16x4_f32(false, a0[k], false, b,
                                                         (short)0, acc0, false, false);
            acc1 = __builtin_amdgcn_wmma_f32_16x16x4_f32(false, a1[k], false, b,
                                                         (short)0, acc1, false, false);
        }
        const float en  = eNormLds[colBase + m];
        const int   cid = colBase + m;
#pragma unroll
        for (int r = 0; r < 8; ++r) {
            // distance (up to row-constant |x|^2) = |e|^2 - 2<x,e>
            float d0 = __builtin_fmaf(-2.0f, acc0[r], en);
            float d1 = __builtin_fmaf(-2.0f, acc1[r], en);
            bool p0 = d0 < minv0[r];   // branchless: EXEC stays all-ones around WMMA
            bool p1 = d1 < minv1[r];
            minv0[r] = p0 ? d0 : minv0[r];  mini0[r] = p0 ? cid : mini0[r];
            minv1[r] = p1 ? d1 : minv1[r];  mini1[r] = p1 ? cid : mini1[r];
        }
    }

    // Butterfly argmin across each 16-lane group (masks < 16 stay in-group).
#pragma unroll
    for (int mask = 8; mask >= 1; mask >>= 1) {
#pragma unroll
        for (int r = 0; r < 8; ++r) {
            float ov0 = __shfl_xor(minv0[r], mask, 32);
            int   oi0 = __shfl_xor(mini0[r], mask, 32);
            float ov1 = __shfl_xor(minv1[r], mask, 32);
            int   oi1 = __shfl_xor(mini1[r], mask, 32);
            bool p0 = (ov0 < minv0[r]) || (ov0 == minv0[r] && oi0 < mini0[r]);
            bool p1 = (ov1 < minv1[r]) || (ov1 == minv1[r] && oi1 < mini1[r]);
            minv0[r] = p0 ? ov0 : minv0[r];  mini0[r] = p0 ? oi0 : mini0[r];
            minv1[r] = p1 ? ov1 : minv1[r];  mini1[r] = p1 ? oi1 : mini1[r];
        }
    }

    if (m == 0) {  // lane 0 -> rows 0..7, lane 16 -> rows 8..15 of each block
#pragma unroll
        for (int r = 0; r < 8; ++r) {
            idxOut[rowBase0 + (hi << 3) + r] = mini0[r];
            idxOut[rowBase1 + (hi << 3) + r] = mini1[r];
        }
    }
}

// ---------------------------------------------------------------------------
// Output kernel: gather + straight-through output + loss. One wave32 per row.
// ---------------------------------------------------------------------------
__global__ __launch_bounds__(256) void vq_output_kernel(const float* __restrict__ h,
                                                        const float* __restrict__ emb,
                                                        const int* __restrict__ idx,
                                                        float* __restrict__ outQ,
                                                        float* __restrict__ outIdx,
                                                        float* __restrict__ outLoss) {
    const int lane = threadIdx.x & 31;
    const int row  = blockIdx.x * (blockDim.x >> 5) + (threadIdx.x >> 5);
    const int id   = idx[row];
    const float x  = h[(size_t)row * EMB_DIM + lane];
    const float q  = emb[(size_t)id * EMB_DIM + lane];
    // quantized_st = h + stop_gradient(q - h): reproduce the exact fp arithmetic
    outQ[(size_t)row * EMB_DIM + lane] = x + (q - x);
    float dsq = (q - x) * (q - x);
#pragma unroll
    for (int mask = 16; mask >= 1; mask >>= 1)
        dsq += __shfl_xor(dsq, mask, 32);
    if (lane == 0) {
        outIdx[row]  = (float)id;
        outLoss[row] = LOSS_SCALE * dsq * (1.0f / EMB_DIM);
    }
}

// ---------------------------------------------------------------------------
extern "C" void kernel_launch(void* const* d_in, const int* in_sizes, int n_in,
                              void* d_out, int out_size, void* d_ws, size_t ws_size,
                              hipStream_t stream) {
    const float* h   = (const float*)d_in[0];
    const float* emb = (const float*)d_in[1];
    const int nVec = in_sizes[0] / EMB_DIM;   // 131072

    int* idxBuf = (int*)d_ws;                 // nVec ints of scratch

    // Output: [quantized_st: nVec*32][indices: nVec][loss: nVec]
    float* outQ    = (float*)d_out;
    float* outIdx  = outQ + (size_t)nVec * EMB_DIM;
    float* outLoss = outIdx + nVec;

    // 32 rows per wave, 8 waves per block -> 256 rows per 256-thread block
    const size_t ldsBytes = (size_t)LDS_FLOATS * sizeof(float);  // ~140 KB of 320 KB/WGP
    vq_argmin_kernel<<<nVec / 256, 256, ldsBytes, stream>>>(h, emb, idxBuf);

    // 1 row per wave, 8 rows per block
    vq_output_kernel<<<nVec / 8, 256, 0, stream>>>(h, emb, idxBuf, outQ, outIdx, outLoss);
}